// Diffusion_59760174956877
// MI455X (gfx1250) — compile-verified
//
#include <hip/hip_runtime.h>
#include <hip/hip_bf16.h>

typedef float v2f __attribute__((ext_vector_type(2)));
typedef float v8f __attribute__((ext_vector_type(8)));

#define L_ATOMS 128
#define R_ATOMS 1024
#define T_HYP   16
#define E_STEPS 32
#define F_DIM   64
#define LR          (L_ATOMS * R_ATOMS)      // 131072
#define FEAT_STRIDE (E_STEPS * F_DIM)        // 2048 floats per atom row

// ---------------------------------------------------------------------------
// Phase 1: atn[e][l][r] = sum_f lig[l,e,f] * rec[r,e,f]
// 32 GEMMs of (128x64)x(64x1024), f32 via V_WMMA_F32_16X16X4_F32.
// One wave32 per 16x16 output tile; K=64 -> 16 WMMA ops per tile.
// A layout (16x4 f32): lane%16 = M row; VGPR v holds K = v + 2*(lane/16).
// B layout (4x16 f32): lane%16 = N col; VGPR v holds K = v + 2*(lane/16).
// C/D layout (16x16 f32): VGPR v holds row M = v + 8*(lane/16), col N = lane%16.
// ---------------------------------------------------------------------------
__global__ __launch_bounds__(256) void atn_wmma_kernel(
    const float* __restrict__ lig,   // [L, E, F]
    const float* __restrict__ rec,   // [R, E, F]
    float* __restrict__ atn)         // [E, L, R]
{
    const int wave = (blockIdx.x * 256 + (int)threadIdx.x) >> 5;  // global wave id
    const int lane = threadIdx.x & 31;

    // 32 e-slices * (8 x 64) tiles each = 16384 waves total
    const int e  = wave >> 9;        // / 512
    const int tt = wave & 511;
    const int tm = tt >> 6;          // tile row   (0..7)
    const int tn = tt & 63;          // tile col   (0..63)

    const int row = lane & 15;
    const int hi  = lane >> 4;       // 0 or 1

    // each lane loads an aligned float2 (K = kk+2*hi, kk+2*hi+1)
    const float* Abase = lig + (tm * 16 + row) * FEAT_STRIDE + e * F_DIM + 2 * hi;
    const float* Bbase = rec + (tn * 16 + row) * FEAT_STRIDE + e * F_DIM + 2 * hi;

    v8f c = {};
#pragma unroll
    for (int kk = 0; kk < F_DIM; kk += 4) {
        v2f a = *reinterpret_cast<const v2f*>(Abase + kk);
        v2f b = *reinterpret_cast<const v2f*>(Bbase + kk);
        // 8 args: (neg_a, A, neg_b, B, c_mod, C, reuse_a, reuse_b)
        c = __builtin_amdgcn_wmma_f32_16x16x4_f32(
                false, a, false, b, (short)0, c, false, false);
    }

    float* Cbase = atn + e * LR + (tm * 16 + 8 * hi) * R_ATOMS + tn * 16 + row;
#pragma unroll
    for (int v = 0; v < 8; ++v)
        Cbase[v * R_ATOMS] = c[v];
}

// ---------------------------------------------------------------------------
// Zero the 16 output energies (harness poisons d_out with 0xAA).
// ---------------------------------------------------------------------------
__global__ void zero_out_kernel(float* __restrict__ out)
{
    if (threadIdx.x < T_HYP) out[threadIdx.x] = 0.0f;
}

// ---------------------------------------------------------------------------
// Phase 2: Us[t] = sum_{l,r,e} atn[e,l,r] * exp(-((d_tlr - mu_e)/sigma)^2)
// One thread per (t, l, r). 32-step unrolled exp loop (v_exp_f32),
// coalesced atn reads (consecutive r across lanes), LDS block reduction,
// one atomicAdd per block.
// ---------------------------------------------------------------------------
__global__ __launch_bounds__(256) void energy_kernel(
    const float* __restrict__ atn,        // [E, L, R]
    const float* __restrict__ lig_coords, // [T, L, 3]
    const float* __restrict__ rec_coords, // [R, 3]
    float* __restrict__ out)              // [T]
{
    const int b   = blockIdx.x;
    const int t   = b >> 9;               // / (L_ATOMS * 4)
    const int rem = b & 511;
    const int l   = rem >> 2;
    const int rc  = rem & 3;
    const int r   = rc * 256 + (int)threadIdx.x;

    // uniform per block -> scalar loads
    const float lx = lig_coords[(t * L_ATOMS + l) * 3 + 0];
    const float ly = lig_coords[(t * L_ATOMS + l) * 3 + 1];
    const float lz = lig_coords[(t * L_ATOMS + l) * 3 + 2];

    const float dx = lx - rec_coords[r * 3 + 0];
    const float dy = ly - rec_coords[r * 3 + 1];
    const float dz = lz - rec_coords[r * 3 + 2];
    const float d  = sqrtf(dx * dx + dy * dy + dz * dz);

    const float inv_sigma = 32.0f / 10.0f;      // |1/sigma| = 3.2 (sign irrelevant, squared)
    const float mu_step   = 10.0f / 31.0f;      // linspace(0,10,32) step

    float acc = 0.0f;
    const float* ap = atn + l * R_ATOMS + r;
#pragma unroll
    for (int e = 0; e < E_STEPS; ++e) {
        float u = (d - (float)e * mu_step) * inv_sigma;
        acc += ap[e * LR] * __expf(-u * u);
    }

    __shared__ float sdata[256];
    sdata[threadIdx.x] = acc;
    __syncthreads();
#pragma unroll
    for (int s = 128; s > 0; s >>= 1) {
        if ((int)threadIdx.x < s) sdata[threadIdx.x] += sdata[threadIdx.x + s];
        __syncthreads();
    }
    if (threadIdx.x == 0)
        atomicAdd(&out[t], sdata[0]);   // ENERGY_SCALE == 1.0
}

// ---------------------------------------------------------------------------
extern "C" void kernel_launch(void* const* d_in, const int* in_sizes, int n_in,
                              void* d_out, int out_size, void* d_ws, size_t ws_size,
                              hipStream_t stream)
{
    const float* lig_feat   = (const float*)d_in[0];  // [128, 32, 64]
    const float* rec_feat   = (const float*)d_in[1];  // [1024, 32, 64]
    const float* lig_coords = (const float*)d_in[2];  // [16, 128, 3]
    const float* rec_coords = (const float*)d_in[3];  // [1024, 3]
    float* out = (float*)d_out;                       // [16]
    float* atn = (float*)d_ws;                        // needs 32*128*1024*4 = 16 MB

    // Phase 1: 16384 tile-waves / 8 waves per block = 2048 blocks
    atn_wmma_kernel<<<2048, 256, 0, stream>>>(lig_feat, rec_feat, atn);

    zero_out_kernel<<<1, 32, 0, stream>>>(out);

    // Phase 2: T * L * (R/256) = 8192 blocks
    energy_kernel<<<T_HYP * L_ATOMS * (R_ATOMS / 256), 256, 0, stream>>>(
        atn, lig_coords, rec_coords, out);
}